// GIN_54898271977857
// MI455X (gfx1250) — compile-verified
//
#include <hip/hip_runtime.h>

// GIN 2-layer forward for MI455X (gfx1250, wave32).
//
// Roofline: edge gather+scatter ~3.3 GB + node traffic ~0.6 GB => ~170us at
// 23.3 TB/s HBM. GEMMs total 6.6 GFLOP (negligible at any matrix rate), so we
// keep them in exact f32 via V_WMMA_F32_16X16X4_F32 — low-precision conversion
// buys nothing on a bandwidth-bound op and would cost accuracy.

#define D 128           // feature dim for all layers
#define WAVE 32

typedef __attribute__((ext_vector_type(2))) float v2f;
typedef __attribute__((ext_vector_type(8))) float v8f;

// ---------------------------------------------------------------------------
// Zero a float buffer (ws is poisoned 0xAA by the harness; must clear agg).
// ---------------------------------------------------------------------------
__global__ void gin_zero_f4(float4* __restrict__ p, long n4) {
  long i = (long)blockIdx.x * blockDim.x + threadIdx.x;
  long stride = (long)gridDim.x * blockDim.x;
  for (; i < n4; i += stride) p[i] = make_float4(0.f, 0.f, 0.f, 0.f);
}

// ---------------------------------------------------------------------------
// Edge scatter-add: agg[dst[e]][:] += feat[src[e]][:]
// One wave per edge; each lane owns 4 consecutive floats (float4 gather is a
// fully coalesced 512B row read; 4x global_atomic_add_f32 per lane, no-return
// so they ride STOREcnt and never stall the wave).
// ---------------------------------------------------------------------------
__global__ __launch_bounds__(256)
void gin_scatter(const float* __restrict__ feat,
                 const int* __restrict__ src,
                 const int* __restrict__ dst,
                 float* __restrict__ agg, int nedges) {
  int wave = (int)((blockIdx.x * (unsigned)blockDim.x + threadIdx.x) >> 5);
  int lane = threadIdx.x & (WAVE - 1);
  if (wave >= nedges) return;

  int s = src[wave];
  int d = dst[wave];

  const float4 v = *(const float4*)(feat + (size_t)s * D + lane * 4);
  float* ap = agg + (size_t)d * D + lane * 4;
  atomicAdd(ap + 0, v.x);
  atomicAdd(ap + 1, v.y);
  atomicAdd(ap + 2, v.z);
  atomicAdd(ap + 3, v.w);
}

// ---------------------------------------------------------------------------
// OUT = act((X + AGG) @ W + bias), W stored [K=128, Ncols=128] (matches the
// reference's torch-Linear-as-[in,out] layout).
//
// Block = 256 threads = 8 waves. Block b computes rows [16b, 16b+16);
// wave w computes the 16x16 output tile at columns [16w, 16w+16).
// K=128 is consumed 4 at a time by V_WMMA_F32_16X16X4_F32 (32 WMMAs/wave).
//
// VGPR layouts (ISA 7.12.2, wave32):
//   A 16x4 : lane&15 = M,  (lane>>4)*2+{0,1} = K  -> v2f {a.x,a.y}
//   B 4x16 : lane&15 = N,  (lane>>4)*2+{0,1} = K  -> v2f {b.x,b.y}
//   C 16x16: VGPR r -> M = r (lanes 0-15) / r+8 (lanes 16-31), N = lane&15
//
// RELU / TAIL_CHECK are compile-time so the epilogue of the common path is
// 8 straight global_store_b32 (no exec-mask churn, no cndmask). EXEC stays
// all-ones across the WMMA loop in both instantiations (clamp via min).
// ---------------------------------------------------------------------------
template <bool RELU, bool TAIL_CHECK>
__global__ __launch_bounds__(256)
void gin_gemm_wmma(const float* __restrict__ X, const float* __restrict__ AGG,
                   const float* __restrict__ W, const float* __restrict__ Bias,
                   float* __restrict__ OUT, int nrows) {
  const int lane = threadIdx.x & (WAVE - 1);
  const int wv   = threadIdx.x >> 5;           // 0..7 -> column tile
  const int m    = lane & 15;
  const int koff = (lane >> 4) << 1;           // 0 or 2
  const int col  = wv * 16 + m;

  const size_t row_base = (size_t)blockIdx.x * 16;
  size_t row = row_base + (size_t)m;
  if (TAIL_CHECK && row >= (size_t)nrows) row = (size_t)nrows - 1;  // keep EXEC full

  const float* __restrict__ xr = X   + row * D;
  const float* __restrict__ ar = AGG + row * D;

  v8f c = {};
#pragma unroll 4
  for (int kk = 0; kk < D; kk += 4) {
    const int k0 = kk + koff;
    v2f a, b;
    a.x = xr[k0]     + ar[k0];
    a.y = xr[k0 + 1] + ar[k0 + 1];
    b.x = W[(size_t)k0 * D + col];
    b.y = W[(size_t)(k0 + 1) * D + col];
    // D(16x16,f32) = A(16x4,f32) x B(4x16,f32) + C
    c = __builtin_amdgcn_wmma_f32_16x16x4_f32(
        /*neg_a=*/false, a, /*neg_b=*/false, b,
        /*c_mod=*/(short)0, c, /*reuse_a=*/false, /*reuse_b=*/false);
  }

  const float bias = Bias[col];
  float* __restrict__ op = OUT + (row_base + (size_t)((lane < 16) ? 0 : 8)) * D + col;
#pragma unroll
  for (int r = 0; r < 8; ++r) {
    float v = c[r] + bias;
    if (RELU) v = fmaxf(v, 0.0f);
    if (TAIL_CHECK) {
      const size_t orow = row_base + (size_t)((lane < 16) ? r : r + 8);
      if (orow < (size_t)nrows) OUT[orow * D + col] = v;
    } else {
      op[(size_t)r * D] = v;
    }
  }
}

// ---------------------------------------------------------------------------
// Host-side launcher (graph-capture safe: kernels only, all on `stream`).
// ---------------------------------------------------------------------------
extern "C" void kernel_launch(void* const* d_in, const int* in_sizes, int n_in,
                              void* d_out, int out_size, void* d_ws, size_t ws_size,
                              hipStream_t stream) {
  const float* x   = (const float*)d_in[0];
  const int*   ei  = (const int*)  d_in[1];   // edge_index [2, E]
  const float* W1  = (const float*)d_in[2];
  const float* b1  = (const float*)d_in[3];
  const float* W2  = (const float*)d_in[4];
  const float* b2  = (const float*)d_in[5];
  float*       out = (float*)d_out;

  const int nnodes = in_sizes[0] / D;
  const int nedges = in_sizes[1] / 2;
  const int* src = ei;
  const int* dst = ei + nedges;

  // Workspace layout: agg [N*D] f32 (reused for both layers), h [N*D] f32.
  float* agg = (float*)d_ws;
  float* h   = agg + (size_t)nnodes * D;

  const long n4 = (long)nnodes * D / 4;
  const dim3 zgrid(2048), blk(256);
  const dim3 sgrid((unsigned)((nedges + 7) / 8));          // 8 edges / block
  const dim3 ggrid((unsigned)((nnodes + 15) / 16));        // 16 rows / block
  const bool full = (nnodes % 16) == 0;                    // true: N=100000

  // ---- Layer 1: agg = scatter(x); h = relu((x+agg)@W1 + b1) ----
  gin_zero_f4<<<zgrid, blk, 0, stream>>>((float4*)agg, n4);
  gin_scatter<<<sgrid, blk, 0, stream>>>(x, src, dst, agg, nedges);
  if (full)
    gin_gemm_wmma<true, false><<<ggrid, blk, 0, stream>>>(x, agg, W1, b1, h, nnodes);
  else
    gin_gemm_wmma<true, true ><<<ggrid, blk, 0, stream>>>(x, agg, W1, b1, h, nnodes);

  // ---- Layer 2: agg = scatter(h); out = (h+agg)@W2 + b2 ----
  gin_zero_f4<<<zgrid, blk, 0, stream>>>((float4*)agg, n4);
  gin_scatter<<<sgrid, blk, 0, stream>>>(h, src, dst, agg, nedges);
  if (full)
    gin_gemm_wmma<false, false><<<ggrid, blk, 0, stream>>>(h, agg, W2, b2, out, nnodes);
  else
    gin_gemm_wmma<false, true ><<<ggrid, blk, 0, stream>>>(h, agg, W2, b2, out, nnodes);
}